// SingleMambaBlock_7834020347951
// MI455X (gfx1250) — compile-verified
//
#include <hip/hip_runtime.h>
#include <cmath>

#define DIMC   256
#define L_SEQ  4096
#define BATCH  8
#define NSTATE 4
#define DTRANK 16
#define DCONV  4

typedef __attribute__((ext_vector_type(2))) float v2f;
typedef __attribute__((ext_vector_type(8))) float v8f;

__device__ __forceinline__ float siluf(float x) { return x / (1.0f + __expf(-x)); }
__device__ __forceinline__ float softplusf(float x) { return (x > 20.0f) ? x : log1pf(__expf(x)); }

// ---------------------------------------------------------------------------
// K1: input layernorm.  x is (B, C, L); h out is (B*L, C) row-major.
// LDS transpose (pad 1 keeps column reads bank-conflict-free).
// ---------------------------------------------------------------------------
#define TILE_L 32
__global__ void k_ln_in(const float* __restrict__ x, const float* __restrict__ w,
                        const float* __restrict__ bia, float* __restrict__ h) {
    __shared__ float tile[DIMC][TILE_L + 1];
    const int tid  = threadIdx.x;                      // 256 threads
    const size_t row0 = (size_t)blockIdx.x * TILE_L;   // global row = b*L + l
    const size_t b  = row0 >> 12;
    const size_t l0 = row0 & (L_SEQ - 1);

    for (int i = 0; i < (DIMC * TILE_L) / 256; ++i) {
        int idx = i * 256 + tid;
        int lt  = idx & (TILE_L - 1);
        int c   = idx >> 5;
        tile[c][lt] = x[(b * DIMC + c) * (size_t)L_SEQ + l0 + lt];
    }
    __syncthreads();

    const int wave = tid >> 5, lane = tid & 31;
    for (int r = 0; r < TILE_L / 8; ++r) {             // 8 waves, 4 rows each
        int lt = wave * (TILE_L / 8) + r;
        float s = 0.f, s2 = 0.f;
        for (int c = lane; c < DIMC; c += 32) {
            float v = tile[c][lt];
            s += v; s2 += v * v;
        }
        for (int off = 16; off >= 1; off >>= 1) {
            s  += __shfl_xor(s,  off, 32);
            s2 += __shfl_xor(s2, off, 32);
        }
        float mu   = s * (1.0f / DIMC);
        float var  = s2 * (1.0f / DIMC) - mu * mu;
        float rstd = rsqrtf(var + 1e-5f);
        size_t orow = (row0 + lt) * DIMC;
        for (int c = lane; c < DIMC; c += 32)
            h[orow + c] = (tile[c][lt] - mu) * rstd * w[c] + bia[c];
    }
}

// ---------------------------------------------------------------------------
// Shared WMMA f32 16x16x4 GEMM core. 64x64 block tile, 16 waves (4x4),
// K step 32. Both LDS tiles row-major along K, pad 4 -> float4 staging,
// conflict-free WMMA column reads.
// ---------------------------------------------------------------------------
#define BM 64
#define BN 64
#define BK 32
#define KPAD 4

__device__ __forceinline__ v8f gemm_tile_64x64(
        const float* __restrict__ A, const float* __restrict__ W,
        float (*As)[BK + KPAD], float (*Bs)[BK + KPAD],
        int m0, int n0, int tid) {
    const int wave = tid >> 5, lane = tid & 31;
    const int wm = wave >> 2, wn = wave & 3;
    const int half = lane >> 4;
    const int mr = wm * 16 + (lane & 15);
    const int nc = wn * 16 + (lane & 15);
    v8f acc = {0.f, 0.f, 0.f, 0.f, 0.f, 0.f, 0.f, 0.f};

    const int r  = tid >> 3;          // 0..63 : tile row
    const int k4 = (tid & 7) * 4;     // float4 within the 32-wide K slice

    for (int k0 = 0; k0 < DIMC; k0 += BK) {
        // A tile: 64x32 floats = 512 float4, one per thread (b128 ld/st)
        *(float4*)&As[r][k4] =
            *(const float4*)&A[(size_t)(m0 + r) * DIMC + k0 + k4];
        // B tile (same orientation as W rows): Bs[nn][kk] = W[n0+nn][k0+kk]
        *(float4*)&Bs[r][k4] =
            *(const float4*)&W[(size_t)(n0 + r) * DIMC + k0 + k4];
        __syncthreads();
        #pragma unroll
        for (int kk = 0; kk < BK; kk += 4) {
            v2f a, bv;
            a.x  = As[mr][kk + 2 * half + 0];
            a.y  = As[mr][kk + 2 * half + 1];
            bv.x = Bs[nc][kk + 2 * half + 0];
            bv.y = Bs[nc][kk + 2 * half + 1];
            acc = __builtin_amdgcn_wmma_f32_16x16x4_f32(
                false, a, false, bv, (short)0, acc, false, false);
        }
        __syncthreads();
    }
    return acc;
}

// ---------------------------------------------------------------------------
// K2: in_proj GEMM  xz = h @ W^T,  W is (512,256).  u = xz[:, :256], z = rest.
// Whole block is either u or z (n0 multiple of 64), so branch is hoisted.
// ---------------------------------------------------------------------------
__global__ void k_gemm_inproj(const float* __restrict__ A, const float* __restrict__ W,
                              float* __restrict__ u, float* __restrict__ z) {
    __shared__ float As[BM][BK + KPAD];
    __shared__ float Bs[BN][BK + KPAD];
    const int tid = threadIdx.x;                  // 512 threads = 16 waves
    const int wave = tid >> 5, lane = tid & 31;
    const int wm = wave >> 2, wn = wave & 3;
    const int m0 = blockIdx.y * BM;
    const int n0 = blockIdx.x * BN;

    v8f acc = gemm_tile_64x64(A, W, As, Bs, m0, n0, tid);

    float* dst = (n0 < DIMC) ? u : z;
    const int nbase = (n0 < DIMC) ? n0 : (n0 - DIMC);
    const int rowBase = m0 + wm * 16 + (lane >> 4) * 8;
    const int col     = nbase + wn * 16 + (lane & 15);
    #pragma unroll
    for (int i = 0; i < 8; ++i)
        dst[(size_t)(rowBase + i) * DIMC + col] = acc[i];
}

// ---------------------------------------------------------------------------
// K3: causal depthwise conv (K=4) + SiLU, optionally over flipped sequence.
// ---------------------------------------------------------------------------
__global__ void k_conv_silu(const float* __restrict__ u, const float* __restrict__ cw,
                            const float* __restrict__ cb, float* __restrict__ xc, int flip) {
    size_t idx = (size_t)blockIdx.x * blockDim.x + threadIdx.x;  // B*L*C
    int d = (int)(idx & (DIMC - 1));
    size_t bl = idx >> 8;
    int l = (int)(bl & (L_SEQ - 1));
    size_t b = bl >> 12;
    float acc = cb[d];
    #pragma unroll
    for (int k = 0; k < DCONV; ++k) {
        int lk = l - (DCONV - 1) + k;
        if (lk >= 0) {
            int lsrc = flip ? (L_SEQ - 1 - lk) : lk;
            acc += cw[k * DIMC + d] * u[(b * L_SEQ + lsrc) * DIMC + d];
        }
    }
    xc[idx] = siluf(acc);
}

// ---------------------------------------------------------------------------
// K4: x_proj (256->24) + dt_proj (16->256) + softplus. One row per block.
// ---------------------------------------------------------------------------
__global__ void k_proj(const float* __restrict__ xc, const float* __restrict__ xpw,
                       const float* __restrict__ dtw, const float* __restrict__ dtb,
                       float* __restrict__ dt, float* __restrict__ bc) {
    __shared__ float row[DIMC];
    __shared__ float xdbl[DTRANK + 2 * NSTATE];
    const int tid = threadIdx.x;                  // 256 threads
    const size_t r = blockIdx.x;                  // B*L rows
    row[tid] = xc[r * DIMC + tid];
    __syncthreads();
    if (tid < DTRANK + 2 * NSTATE) {
        float s = 0.f;
        #pragma unroll 8
        for (int c = 0; c < DIMC; ++c) s += xpw[tid * DIMC + c] * row[c];
        xdbl[tid] = s;
    }
    __syncthreads();
    float s = dtb[tid];
    #pragma unroll
    for (int k = 0; k < DTRANK; ++k) s += dtw[tid * DTRANK + k] * xdbl[k];
    dt[r * DIMC + tid] = softplusf(s);
    if (tid < 2 * NSTATE) bc[r * 8 + tid] = xdbl[DTRANK + tid];
}

// ---------------------------------------------------------------------------
// K5: selective scan. One thread per (b,d), n=4 states in registers.
// Coalesced dt/u loads, float4 broadcast B/C loads, prefetch hints.
// ---------------------------------------------------------------------------
__global__ void k_scan(const float* __restrict__ xc, const float* __restrict__ dt,
                       const float* __restrict__ bcbuf, const float* __restrict__ A_log,
                       const float* __restrict__ Dp, float* __restrict__ ys) {
    int tid = blockIdx.x * blockDim.x + threadIdx.x;   // B*DIMC = 2048
    int d = tid & (DIMC - 1);
    size_t b = (size_t)(tid >> 8);
    float a0 = -__expf(A_log[d * NSTATE + 0]);
    float a1 = -__expf(A_log[d * NSTATE + 1]);
    float a2 = -__expf(A_log[d * NSTATE + 2]);
    float a3 = -__expf(A_log[d * NSTATE + 3]);
    float Dd = Dp[d];
    float h0 = 0.f, h1 = 0.f, h2 = 0.f, h3 = 0.f;
    const float* dtp = dt + (b * L_SEQ) * DIMC + d;
    const float* up  = xc + (b * L_SEQ) * DIMC + d;
    const float4* bp = (const float4*)(bcbuf + (b * L_SEQ) * 8);
    float* yp = ys + (b * L_SEQ) * DIMC + d;
    #pragma unroll 2
    for (int t = 0; t < L_SEQ; ++t) {
        if (t + 16 < L_SEQ) {
            __builtin_prefetch(dtp + (size_t)(t + 16) * DIMC, 0, 0);
            __builtin_prefetch(up  + (size_t)(t + 16) * DIMC, 0, 0);
        }
        float dtv = dtp[(size_t)t * DIMC];
        float uv  = up[(size_t)t * DIMC];
        float du  = dtv * uv;
        float4 Bv = bp[t * 2 + 0];
        float4 Cv = bp[t * 2 + 1];
        h0 = __expf(dtv * a0) * h0 + du * Bv.x;
        h1 = __expf(dtv * a1) * h1 + du * Bv.y;
        h2 = __expf(dtv * a2) * h2 + du * Bv.z;
        h3 = __expf(dtv * a3) * h3 + du * Bv.w;
        float y = h0 * Cv.x + h1 * Cv.y + h2 * Cv.z + h3 * Cv.w;
        yp[(size_t)t * DIMC] = y + uv * Dd;
    }
}

// ---------------------------------------------------------------------------
// K6: gate + combine directions + output layernorm. One wave per row,
// each lane owns 8 contiguous d (float4 x2 loads/stores).
// ---------------------------------------------------------------------------
__global__ void k_comb_ln(const float* __restrict__ ysf, const float* __restrict__ ysb,
                          const float* __restrict__ z, const float* __restrict__ w,
                          const float* __restrict__ bia, float* __restrict__ yn) {
    const int wave = threadIdx.x >> 5, lane = threadIdx.x & 31;
    const size_t row = (size_t)blockIdx.x * 8 + wave;  // B*L rows
    const int l = (int)(row & (L_SEQ - 1));
    const size_t b = row >> 12;
    const size_t rb = b * L_SEQ + (L_SEQ - 1 - l);
    const int d0 = lane * 8;
    float v[8];
    float s = 0.f, s2 = 0.f;
    #pragma unroll
    for (int q = 0; q < 2; ++q) {
        float4 zf = *(const float4*)&z[row * DIMC + d0 + 4 * q];
        float4 yf = *(const float4*)&ysf[row * DIMC + d0 + 4 * q];
        float4 yb = *(const float4*)&ysb[rb * DIMC + d0 + 4 * q];
        float zz[4] = {zf.x, zf.y, zf.z, zf.w};
        float ff[4] = {yf.x, yf.y, yf.z, yf.w};
        float bb[4] = {yb.x, yb.y, yb.z, yb.w};
        #pragma unroll
        for (int j = 0; j < 4; ++j) {
            float val = 0.5f * siluf(zz[j]) * (ff[j] + bb[j]);
            v[4 * q + j] = val; s += val; s2 += val * val;
        }
    }
    for (int off = 16; off >= 1; off >>= 1) {
        s  += __shfl_xor(s,  off, 32);
        s2 += __shfl_xor(s2, off, 32);
    }
    float mu   = s * (1.0f / DIMC);
    float var  = s2 * (1.0f / DIMC) - mu * mu;
    float rstd = rsqrtf(var + 1e-5f);
    #pragma unroll
    for (int q = 0; q < 2; ++q) {
        float4 wf = *(const float4*)&w[d0 + 4 * q];
        float4 bf = *(const float4*)&bia[d0 + 4 * q];
        float4 o;
        o.x = (v[4 * q + 0] - mu) * rstd * wf.x + bf.x;
        o.y = (v[4 * q + 1] - mu) * rstd * wf.y + bf.y;
        o.z = (v[4 * q + 2] - mu) * rstd * wf.z + bf.z;
        o.w = (v[4 * q + 3] - mu) * rstd * wf.w + bf.w;
        *(float4*)&yn[row * DIMC + d0 + 4 * q] = o;
    }
}

// ---------------------------------------------------------------------------
// K7: out_proj GEMM + transposed residual store to (B, C, L).
// C-tile transposed through LDS; final store is float4 along L.
// ---------------------------------------------------------------------------
__global__ void k_gemm_out(const float* __restrict__ A, const float* __restrict__ W,
                           const float* __restrict__ xin, float* __restrict__ out) {
    __shared__ float As[BM][BK + KPAD];
    __shared__ float Bs[BN][BK + KPAD];
    __shared__ float Ct[BN][BM + KPAD];
    const int tid = threadIdx.x;                  // 512 threads = 16 waves
    const int wave = tid >> 5, lane = tid & 31;
    const int wm = wave >> 2, wn = wave & 3;
    const int m0 = blockIdx.y * BM;
    const int n0 = blockIdx.x * BN;

    v8f acc = gemm_tile_64x64(A, W, As, Bs, m0, n0, tid);

    // stage transposed C tile (each lane: 8 contiguous floats = 2 float4)
    const int mmBase = wm * 16 + (lane >> 4) * 8;
    const int nnLoc  = wn * 16 + (lane & 15);
    {
        float4 c0 = make_float4(acc[0], acc[1], acc[2], acc[3]);
        float4 c1 = make_float4(acc[4], acc[5], acc[6], acc[7]);
        *(float4*)&Ct[nnLoc][mmBase + 0] = c0;
        *(float4*)&Ct[nnLoc][mmBase + 4] = c1;
    }
    __syncthreads();
    // coalesced transposed store with residual: out[b, c, l] = C^T + x
    #pragma unroll
    for (int it = 0; it < (BN * BM) / (512 * 4); ++it) {   // 2 iters
        int idx = it * 512 + tid;
        int nn = idx >> 4;            // 16 float4-groups per 64-wide row
        int m4 = (idx & 15) * 4;
        size_t rowb = (size_t)m0 + m4;
        int l = (int)(rowb & (L_SEQ - 1));
        size_t b = rowb >> 12;
        size_t o = (b * DIMC + (size_t)(n0 + nn)) * L_SEQ + l;
        float4 cv = *(const float4*)&Ct[nn][m4];
        float4 rv = *(const float4*)&xin[o];
        cv.x += rv.x; cv.y += rv.y; cv.z += rv.z; cv.w += rv.w;
        *(float4*)&out[o] = cv;
    }
}

// ---------------------------------------------------------------------------
extern "C" void kernel_launch(void* const* d_in, const int* in_sizes, int n_in,
                              void* d_out, int out_size, void* d_ws, size_t ws_size,
                              hipStream_t stream) {
    const float* x         = (const float*)d_in[0];
    const float* ln_w      = (const float*)d_in[1];
    const float* ln_b      = (const float*)d_in[2];
    const float* in_proj_w = (const float*)d_in[3];
    const float* conv_w_f  = (const float*)d_in[4];
    const float* conv_b_f  = (const float*)d_in[5];
    const float* xpw_f     = (const float*)d_in[6];
    const float* dtw_f     = (const float*)d_in[7];
    const float* dtb_f     = (const float*)d_in[8];
    const float* A_log_f   = (const float*)d_in[9];
    const float* D_f       = (const float*)d_in[10];
    const float* conv_w_b  = (const float*)d_in[11];
    const float* conv_b_b  = (const float*)d_in[12];
    const float* xpw_b     = (const float*)d_in[13];
    const float* dtw_b     = (const float*)d_in[14];
    const float* dtb_b     = (const float*)d_in[15];
    const float* A_log_b   = (const float*)d_in[16];
    const float* D_b       = (const float*)d_in[17];
    const float* nrm_w     = (const float*)d_in[18];
    const float* nrm_b     = (const float*)d_in[19];
    const float* out_w     = (const float*)d_in[20];
    float* out = (float*)d_out;

    const size_t NL  = (size_t)BATCH * L_SEQ;       // 32768 rows
    const size_t NLd = NL * DIMC;                   // floats per (B,L,C) buffer
    float* ws  = (float*)d_ws;
    float* h   = ws;                                // slot 0 (reused as dt)
    float* u   = ws + 1 * NLd;
    float* z   = ws + 2 * NLd;
    float* xc  = ws + 3 * NLd;                      // reused as yn
    float* ysf = ws + 4 * NLd;
    float* ysb = ws + 5 * NLd;
    float* bc  = ws + 6 * NLd;                      // NL*8 floats
    float* dt_buf = h;                              // h dead after in_proj GEMM
    float* yn  = xc;                                // xc dead after both scans

    // 1) input layernorm -> h (B*L, 256)
    k_ln_in<<<(int)(NL / TILE_L), 256, 0, stream>>>(x, ln_w, ln_b, h);
    // 2) in_proj GEMM -> u, z
    k_gemm_inproj<<<dim3((2 * DIMC) / BN, (int)(NL / BM)), 512, 0, stream>>>(h, in_proj_w, u, z);
    // 3) forward direction
    k_conv_silu<<<(int)(NLd / 256), 256, 0, stream>>>(u, conv_w_f, conv_b_f, xc, 0);
    k_proj<<<(int)NL, 256, 0, stream>>>(xc, xpw_f, dtw_f, dtb_f, dt_buf, bc);
    k_scan<<<(BATCH * DIMC) / 256, 256, 0, stream>>>(xc, dt_buf, bc, A_log_f, D_f, ysf);
    // 4) backward direction (flipped)
    k_conv_silu<<<(int)(NLd / 256), 256, 0, stream>>>(u, conv_w_b, conv_b_b, xc, 1);
    k_proj<<<(int)NL, 256, 0, stream>>>(xc, xpw_b, dtw_b, dtb_b, dt_buf, bc);
    k_scan<<<(BATCH * DIMC) / 256, 256, 0, stream>>>(xc, dt_buf, bc, A_log_b, D_b, ysb);
    // 5) gate + combine + output layernorm -> yn
    k_comb_ln<<<(int)(NL / 8), 256, 0, stream>>>(ysf, ysb, z, nrm_w, nrm_b, yn);
    // 6) out_proj GEMM + residual, transposed store to (B, C, L)
    k_gemm_out<<<dim3(DIMC / BN, (int)(NL / BM)), 512, 0, stream>>>(yn, out_w, x, out);
}